// Ect2DPointsLayer_33621003993806
// MI455X (gfx1250) — compile-verified
//
#include <hip/hip_runtime.h>

typedef __attribute__((ext_vector_type(2))) float v2f;
typedef __attribute__((ext_vector_type(8))) float v8f;

#define T_DIM   64
#define S_DIM   64
#define S_CHUNK 16
#define BIG     1.0e9f

#if defined(__has_builtin)
#  if __has_builtin(__builtin_amdgcn_tanhf)
#    define HAVE_TANH 1
#  endif
#endif

#ifdef HAVE_TANH
#  define SCALE 100.0f               /* y/2 = 100*(lin - nh) for tanh form */
#else
#  define SCALE 288.5390081777927f   /* 200 * log2(e) for exp2 form */
#endif

__device__ __forceinline__ int lower_bound_i32(const int* __restrict__ a, int n, int key) {
    int lo = 0, hi = n;
    while (lo < hi) {
        int mid = (lo + hi) >> 1;
        if (a[mid] < key) lo = mid + 1; else hi = mid;
    }
    return lo;
}

__global__ __launch_bounds__(256)
void ect2d_wmma_kernel(const float* __restrict__ x,
                       const float* __restrict__ v,
                       const float* __restrict__ lin,
                       const int*   __restrict__ batch,
                       float*       __restrict__ out,
                       int n) {
    const int b    = blockIdx.y;            // graph id
    const int s0   = blockIdx.x * S_CHUNK;  // filtration-height chunk
    const int tid  = threadIdx.x;
    const int lane = tid & 31;
    const int wave = tid >> 5;              // 0..7
    const int half = lane >> 4;             // 0: lanes 0-15, 1: lanes 16-31
    const int col  = lane & 15;
    const int t    = (wave & 3) * 16 + col; // theta owned by this lane

    __shared__ float tile[S_CHUNK * T_DIM]; // [s][t] block-local accumulator
    __shared__ int   seg[2];

    for (int i = tid; i < S_CHUNK * T_DIM; i += 256) tile[i] = 0.0f;
    if (tid == 0) {
        seg[0] = lower_bound_i32(batch, n, b);
        seg[1] = lower_bound_i32(batch, n, b + 1);
    }
    __syncthreads();
    const int start = seg[0];
    const int end   = seg[1];

    // ---- B matrix: direction vectors, dual-layout hedge (K1 slot covered twice,
    // A's K2/K3 rows are zero so the redundant slot is multiplied by 0).
    const float vx = v[2 * t + 0];
    const float vy = v[2 * t + 1];
    v2f bm;
    bm.x = half ? vy : vx;   // VGPR0: K0 = v.x (lanes 0-15), upper half = v.y
    bm.y = half ? 0.0f : vy; // VGPR1: K1 = v.y (lanes 0-15), upper half = 0

    // ---- precompute SCALE*lin[s] for this s-chunk (hoisted to SGPRs)
    float a2[S_CHUNK];
    #pragma unroll
    for (int s = 0; s < S_CHUNK; ++s) a2[s] = SCALE * lin[s0 + s];

    float acc[S_CHUNK];
    #pragma unroll
    for (int s = 0; s < S_CHUNK; ++s) acc[s] = 0.0f;

    float iters = 0.0f;   // loop-trip counter (uniform per wave) for tanh affine fixup

    // waves 0-3 take even 16-point chunks, waves 4-7 take odd chunks
    for (int pbase = start + (wave >> 2) * 16; pbase < end; pbase += 32) {
        iters += 1.0f;
        // ---- A matrix: 16 points x (K=2 padded to 4). Rows = lanes 0-15.
        v2f am; am.x = 0.0f; am.y = 0.0f;
        if (half == 0) {
            int p = pbase + col;
            if (p < end) { am.x = x[2 * p]; am.y = x[2 * p + 1]; }
        }
        v8f c = {0.f, 0.f, 0.f, 0.f, 0.f, 0.f, 0.f, 0.f};
        // nh tile: D[point, theta] = x . v   (v_wmma_f32_16x16x4_f32)
        v8f d = __builtin_amdgcn_wmma_f32_16x16x4_f32(
                    false, am, false, bm, (short)0, c, false, false);

        #pragma unroll
        for (int r = 0; r < 8; ++r) {
            int p = pbase + r + 8 * half;            // D row for (VGPR r, lane half)
            float nh = (p < end) ? d[r] : BIG;       // tail mask: contribution -> 0
            float m  = SCALE * nh;
            #pragma unroll
            for (int s = 0; s < S_CHUNK; ++s) {
#ifdef HAVE_TANH
                // sigmoid(200*(lin-nh)) = 0.5 + 0.5*tanh(100*(lin-nh))
                // accumulate raw tanh; affine applied once after the loop.
                // invalid point: tanh(-1e11) = -1  ->  0.5 - 0.5 = 0 exactly.
                acc[s] += __builtin_amdgcn_tanhf(a2[s] - m);
#else
                // sigmoid(200*(lin-nh)) = 1 / (1 + 2^(C1*nh - C1*lin))
                float e = __builtin_amdgcn_exp2f(m - a2[s]);
                acc[s] += __builtin_amdgcn_rcpf(1.0f + e);
#endif
            }
        }
    }

    // ---- combine lanes/waves sharing the same theta via LDS float atomics
#ifdef HAVE_TANH
    const float baseM = 4.0f * iters;   // 0.5 * (8 elements per iter) * iters
#endif
    #pragma unroll
    for (int s = 0; s < S_CHUNK; ++s) {
#ifdef HAVE_TANH
        atomicAdd(&tile[s * T_DIM + t], fmaf(0.5f, acc[s], baseM));
#else
        atomicAdd(&tile[s * T_DIM + t], acc[s]);
#endif
    }
    __syncthreads();

    // ---- block owns out[b, s0:s0+16, :] exclusively: plain coalesced stores
    float* outb = out + (size_t)b * (S_DIM * T_DIM) + (size_t)s0 * T_DIM;
    for (int i = tid; i < S_CHUNK * T_DIM; i += 256)
        outb[i] = tile[i];
}

extern "C" void kernel_launch(void* const* d_in, const int* in_sizes, int n_in,
                              void* d_out, int out_size, void* d_ws, size_t ws_size,
                              hipStream_t stream) {
    (void)n_in; (void)d_ws; (void)ws_size;
    const float* x     = (const float*)d_in[0];   // [N,2] f32
    const float* v     = (const float*)d_in[1];   // [64,2] f32
    const float* lin   = (const float*)d_in[2];   // [64] f32
    const int*   batch = (const int*)d_in[3];     // [N] i32, sorted
    float*       out   = (float*)d_out;           // [128,64,64] f32

    const int n          = in_sizes[0] / 2;
    const int num_graphs = out_size / (S_DIM * T_DIM); // 128

    dim3 grid(S_DIM / S_CHUNK, num_graphs);  // 4 x 128 = 512 blocks
    ect2d_wmma_kernel<<<grid, 256, 0, stream>>>(x, v, lin, batch, out, n);
}